// LagrKANNautoinner_532575944766
// MI455X (gfx1250) — compile-verified
//
#include <hip/hip_runtime.h>

// Problem constants (from reference)
#define S_TOTAL   4096
#define W_WIDTH   64
#define N_NODES_  257
#define BLK       16448        // W_WIDTH * N_NODES per sample
#define T_ELEMS   262144       // S * W
#define PHI_OFF   786432       // 3 * T_ELEMS
#define DPHI_OFF  68157440     // PHI_OFF + S*W*N
#define DDPHI_OFF 135528448
#define DELTA_OFF 202899456

typedef __attribute__((ext_vector_type(2))) float v2f;
typedef __attribute__((ext_vector_type(8))) float v8f;

// Order-4 Lagrange basis + 1st/2nd derivatives at reference coord, with the
// derivative scaling by 1/delta_x = 128 and 1/delta_x^2 = 16384 folded in.
__device__ __forceinline__ void lagr_basis(float x, int& base,
                                           float phi[5], float dphi[5], float ddphi[5]) {
  const float nodes[5] = {-1.0f, -0.5f, 0.0f, 0.5f, 1.0f};
  float xs = 256.0f * x;                       // x_shift
  float fe = floorf(xs * 0.25f);
  fe = fminf(fmaxf(fe, 0.0f), 63.0f);          // clip element id
  base = (int)fe * 4;
  float xt = 0.5f * (xs - (float)base) - 1.0f; // to_ref
#pragma unroll
  for (int j = 0; j < 5; ++j) {
    float p = 1.0f;
#pragma unroll
    for (int m = 0; m < 5; ++m)
      if (m != j) p *= (xt - nodes[m]) / (nodes[j] - nodes[m]);
    phi[j] = p;

    float y = 0.0f;
#pragma unroll
    for (int i = 0; i < 5; ++i)
      if (i != j) {
        float k = 1.0f / (nodes[j] - nodes[i]);
#pragma unroll
        for (int m = 0; m < 5; ++m)
          if (m != i && m != j) k *= (xt - nodes[m]) / (nodes[j] - nodes[m]);
        y += k;
      }
    dphi[j] = y * 128.0f;

    float z = 0.0f;
#pragma unroll
    for (int i = 0; i < 5; ++i)
      if (i != j) {
        float ks = 0.0f;
#pragma unroll
        for (int m = 0; m < 5; ++m)
          if (m != i && m != j) {
            float kp = 1.0f / (nodes[j] - nodes[m]);
#pragma unroll
            for (int n = 0; n < 5; ++n)
              if (n != i && n != j && n != m)
                kp *= (xt - nodes[n]) / (nodes[j] - nodes[n]);
            ks += kp;
          }
        z += ks / (nodes[j] - nodes[i]);
      }
    ddphi[j] = z * 16384.0f;
  }
}

// ---------------------------------------------------------------------------
// Kernel 1: the 808 MB store-bandwidth part (roofline ~35 us @ 23.3 TB/s).
// One workgroup (8 waves) per sample writes the phi/dphi/ddphi_ikp [64 x 257]
// blocks with nontemporal b32 stores (write-once data; keep out of L2).
// Every k-row within a block is identical: zeros except 5 basis values at
// columns [base, base+5).
// ---------------------------------------------------------------------------
__global__ void kan_writer(const float* __restrict__ x, float* __restrict__ out) {
  __shared__ float s_basis[15];
  __shared__ int s_base;
  const int i = blockIdx.x;
  if (threadIdx.x == 0) {
    int base; float ph[5], dph[5], ddph[5];
    lagr_basis(x[i], base, ph, dph, ddph);
#pragma unroll
    for (int j = 0; j < 5; ++j) {
      s_basis[j] = ph[j]; s_basis[5 + j] = dph[j]; s_basis[10 + j] = ddph[j];
    }
    s_base = base;
    if (i == 0) out[DELTA_OFF] = 0.0078125f;  // delta_x
  }
  __syncthreads();

  const int base = s_base;
  const int lane = threadIdx.x & 31;
  const int wave = threadIdx.x >> 5;

  // 192 rows (3 tensors x 64 widths), 8 waves -> 24 rows/wave, 257 f32 each.
  for (int r = wave; r < 192; r += 8) {
    const int tensor = r >> 6;
    const int k      = r & 63;
    const size_t toff = (tensor == 0) ? (size_t)PHI_OFF
                      : (tensor == 1) ? (size_t)DPHI_OFF : (size_t)DDPHI_OFF;
    float* rowp = out + toff + (size_t)i * BLK + (size_t)k * N_NODES_;
    const float* bs = &s_basis[tensor * 5];
    for (int p = lane; p < N_NODES_; p += 32) {
      float v = 0.0f;
      unsigned j = (unsigned)(p - base);
      if (j < 5u) v = bs[j];
      __builtin_nontemporal_store(v, rowp + p);
    }
  }
}

// ---------------------------------------------------------------------------
// Kernel 2: t/dt/ddt = Phi_dense @ weight^T via V_WMMA_F32_16X16X4_F32.
// Grid = (S/16) x 4 one-wave blocks; each block owns one 16x16 output tile
// for all three derivatives -> only 3 v8f accumulators live (no spills).
// A fragments built from register-resident basis values via cndmask selects
// (no LDS / exec-mask traffic in the K loop).
// A layout (ISA 7.12.2, 32-bit A 16x4): M = lane&15, K = 2*(lane>>4) + vgpr.
// B layout (mirror): N = lane&15, K = 2*(lane>>4) + vgpr.
// C/D layout: vgpr v -> row M = v + 8*(lane>>4), col N = lane&15.
// ---------------------------------------------------------------------------
__device__ __forceinline__ float sel5(const float r[5], unsigned j) {
  float v = 0.0f;
  v = (j == 0u) ? r[0] : v;
  v = (j == 1u) ? r[1] : v;
  v = (j == 2u) ? r[2] : v;
  v = (j == 3u) ? r[3] : v;
  v = (j == 4u) ? r[4] : v;
  return v;
}

__global__ void kan_dots_wmma(const float* __restrict__ x,
                              const float* __restrict__ weight,
                              float* __restrict__ out) {
  __shared__ float sb[16][16];  // [sample][0..4]=phi, [5..9]=dphi, [10..14]=ddphi, [15]=base
  const int lane = threadIdx.x;          // 0..31, one wave per block
  const int i0 = (blockIdx.x >> 2) * 16; // sample tile
  const int nt = blockIdx.x & 3;         // width tile

  if (lane < 16) {
    int base; float ph[5], dph[5], ddph[5];
    lagr_basis(x[i0 + lane], base, ph, dph, ddph);
#pragma unroll
    for (int j = 0; j < 5; ++j) {
      sb[lane][j] = ph[j]; sb[lane][5 + j] = dph[j]; sb[lane][10 + j] = ddph[j];
    }
    sb[lane][15] = (float)base;
  }
  __syncthreads();

  const int m    = lane & 15;
  const int half = lane >> 4;

  // Hoist this lane's A-row basis into registers (static indices -> VGPRs).
  float rph[5], rdp[5], rdd[5];
#pragma unroll
  for (int j = 0; j < 5; ++j) {
    rph[j] = sb[m][j];
    rdp[j] = sb[m][5 + j];
    rdd[j] = sb[m][10 + j];
  }
  const int base_m = (int)sb[m][15];

  const int kw = nt * 16 + m;                        // width (B column) index
  const float* wrow = weight + (size_t)kw * N_NODES_;

  v8f acc_t  = {0.f, 0.f, 0.f, 0.f, 0.f, 0.f, 0.f, 0.f};
  v8f acc_d  = acc_t;
  v8f acc_dd = acc_t;

#pragma unroll 4
  for (int c = 0; c < 65; ++c) {                     // K = 260 (257 zero-padded)
    const int p0 = 4 * c + 2 * half;
    const int p1 = p0 + 1;
    const unsigned j0 = (unsigned)(p0 - base_m);
    const unsigned j1 = j0 + 1u;

    v2f a_t  = { sel5(rph, j0), sel5(rph, j1) };
    v2f a_d  = { sel5(rdp, j0), sel5(rdp, j1) };
    v2f a_dd = { sel5(rdd, j0), sel5(rdd, j1) };

    const float b0 = (p0 < N_NODES_) ? wrow[p0] : 0.0f;
    const float b1 = (p1 < N_NODES_) ? wrow[p1] : 0.0f;
    v2f bv = { b0, b1 };

    acc_t  = __builtin_amdgcn_wmma_f32_16x16x4_f32(false, a_t,  false, bv,
                                                   (short)0, acc_t,  false, false);
    acc_d  = __builtin_amdgcn_wmma_f32_16x16x4_f32(false, a_d,  false, bv,
                                                   (short)0, acc_d,  false, false);
    acc_dd = __builtin_amdgcn_wmma_f32_16x16x4_f32(false, a_dd, false, bv,
                                                   (short)0, acc_dd, false, false);
  }

  // Store the 16x16 D tiles: t, dt, ddt are [S, 64] row-major.
  const int col = nt * 16 + m;
#pragma unroll
  for (int v = 0; v < 8; ++v) {
    const int row = i0 + v + 8 * half;
    out[(size_t)row * 64 + col]                       = acc_t[v];
    out[(size_t)T_ELEMS + (size_t)row * 64 + col]     = acc_d[v];
    out[(size_t)2 * T_ELEMS + (size_t)row * 64 + col] = acc_dd[v];
  }
}

extern "C" void kernel_launch(void* const* d_in, const int* in_sizes, int n_in,
                              void* d_out, int out_size, void* d_ws, size_t ws_size,
                              hipStream_t stream) {
  const float* x = (const float*)d_in[0];        // [4096]
  const float* w = (const float*)d_in[1];        // [64, 257]
  float* out = (float*)d_out;                    // 202,899,457 f32
  (void)in_sizes; (void)n_in; (void)out_size; (void)d_ws; (void)ws_size;

  // ~808 MB of NT stores: the bandwidth-bound part
  kan_writer<<<S_TOTAL, 256, 0, stream>>>(x, out);
  // t/dt/ddt GEMMs via WMMA f32 16x16x4 (one wave per 16x16 output tile)
  kan_dots_wmma<<<(S_TOTAL / 16) * 4, 32, 0, stream>>>(x, w, out);
}